// DescCGFLoss_30992484008038
// MI455X (gfx1250) — compile-verified
//
#include <hip/hip_runtime.h>
#include <math.h>

#define BB 2
#define CC 64
#define MM 1024
#define R2 2.25f   // CGF_RADIUS^2

typedef __attribute__((ext_vector_type(2))) float v2f;
typedef __attribute__((ext_vector_type(8))) float v8f;

__device__ __forceinline__ float wave_sum32(float v) {
#pragma unroll
  for (int off = 16; off > 0; off >>= 1) v += __shfl_xor(v, off, 32);
  return v;
}

// wave-local LDS ordering fence: same-wave DS ops are in-order in the LDS
// pipe; this just stops the compiler reordering and drains DScnt.
__device__ __forceinline__ void wave_lds_fence() {
  asm volatile("s_wait_dscnt 0" ::: "memory");
}

// ---------------------------------------------------------------------------
// K1: per-row triplet selection + gathered descriptor distances.
// 4 waves / block, each wave owns a 16-row block. V_WMMA_F32_16X16X4_F32
// computes the exact 16x16 keypoint Gram tile (K=3 padded to 4); VALU does
// the streaming argmax/argmin over the random matrices (bandwidth-bound).
// ---------------------------------------------------------------------------
__global__ __launch_bounds__(128)
void k1_select(const float* __restrict__ akp, const float* __restrict__ adc,
               const float* __restrict__ pkp, const float* __restrict__ pdc,
               const float* __restrict__ rs,
               const float* __restrict__ rm,  const float* __restrict__ ro,
               float* __restrict__ bcl, float* __restrict__ anyv)
{
  __shared__ float s_np[MM];        // pos keypoint squared norms (exact f32)
  __shared__ float s_pk[3][MM];     // pos keypoint coords
  __shared__ float s_na[4][16];     // anchor squared norms per wave
  __shared__ float s_tile[4][256];  // 16x16 dot tile per wave
  __shared__ int   s_idx[4][16][3]; // selected indices per row
  __shared__ float s_any[4][16];    // any-in-radius mask per row

  const int tid  = threadIdx.x;
  const int lane = tid & 31;
  const int wave = tid >> 5;
  const int b    = blockIdx.x >> 4;
  const int i0   = ((blockIdx.x & 15) * 4 + wave) * 16;

  const float* pk = pkp + (size_t)b * 3 * MM;
  const float* ak = akp + (size_t)b * 3 * MM;

  // stage pos keypoints + norms (block-wide, exact f32)
  for (int j = tid; j < MM; j += 128) {
    float x = pk[j];
    float y = pk[MM + j];
    float z = pk[2 * MM + j];
    s_pk[0][j] = x; s_pk[1][j] = y; s_pk[2][j] = z;
    s_np[j] = x * x + y * y + z * z;
  }

  // A fragment: 16x4 f32. Layout: lanes 0-15 hold (M=lane, K=0,1) in
  // VGPR0,VGPR1; lanes 16-31 hold (M=lane-16, K=2,3). K=3 is zero pad.
  const int nh = lane & 15;
  const int mrow = i0 + nh;
  {
    // all lanes load the 3 coords of their row (coalesced 16-wide)
  }
  float axv = ak[mrow];
  float ayv = ak[MM + mrow];
  float azv = ak[2 * MM + mrow];
  v2f a2;
  a2[0] = (lane < 16) ? axv : azv;
  a2[1] = (lane < 16) ? ayv : 0.0f;
  if (lane < 16) s_na[wave][nh] = axv * axv + ayv * ayv + azv * azv;
  __syncthreads();   // the only block-wide barrier (covers s_pk/s_np/s_na)

  const int r = lane >> 1;        // row within 16-row block (2 lanes per row)
  const int h = lane & 1;         // column half (8 cols each)
  const int i_row = i0 + r;
  const float na_r = s_na[wave][r];
  const float* rm_row = rm + ((size_t)b * MM + i_row) * MM;
  const float* ro_row = ro + ((size_t)b * MM + i_row) * MM;

  float bpV = -1.0f;   int bpI = 0;  // argmax(mask * rand)
  float bnV = 3.0e38f; int bnI = 0;  // argmin(kd2 + 1000*mask)
  float boV = -1.0f;   int boI = 0;  // argmax(rand_out * !mask)
  float anyr = 0.0f;

  for (int jt = 0; jt < MM / 16; ++jt) {
    const int j0 = jt * 16;

    // B fragment: 4x16 f32. lanes 0-15: (K=0,1, N=lane); lanes 16-31: (K=2,3).
    const int j = j0 + nh;
    float pxv = s_pk[0][j];
    float pyv = s_pk[1][j];
    float pzv = s_pk[2][j];
    v2f b2;
    b2[0] = (lane < 16) ? pxv : pzv;
    b2[1] = (lane < 16) ? pyv : 0.0f;

    v8f c = {0.f, 0.f, 0.f, 0.f, 0.f, 0.f, 0.f, 0.f};
    c = __builtin_amdgcn_wmma_f32_16x16x4_f32(false, a2, false, b2,
                                              (short)0, c, false, false);
    // C layout: VGPR v, lanes 0-15 -> (M=v, N=lane); lanes 16-31 -> (M=v+8)
    const int halfw = lane >> 4;
#pragma unroll
    for (int v = 0; v < 8; ++v)
      s_tile[wave][(v + 8 * halfw) * 16 + nh] = c[v];
    wave_lds_fence();   // wave-local: tile is private to this wave

    const int jc = j0 + h * 8;
    if (jt + 1 < MM / 16) {
      __builtin_prefetch(rm_row + jc + 16, 0, 3);   // WGP-scope temporal
      __builtin_prefetch(ro_row + jc + 16, 0, 3);
    }
    const float4 m0 = *(const float4*)(rm_row + jc);
    const float4 m1 = *(const float4*)(rm_row + jc + 4);
    const float4 o0 = *(const float4*)(ro_row + jc);
    const float4 o1 = *(const float4*)(ro_row + jc + 4);
    const float mv[8] = {m0.x, m0.y, m0.z, m0.w, m1.x, m1.y, m1.z, m1.w};
    const float ov[8] = {o0.x, o0.y, o0.z, o0.w, o1.x, o1.y, o1.z, o1.w};
#pragma unroll
    for (int k = 0; k < 8; ++k) {
      const int jj = jc + k;
      const float dot = s_tile[wave][r * 16 + h * 8 + k];
      const float kd2 = na_r + s_np[jj] - 2.0f * dot;
      const bool inR = (kd2 <= R2);
      anyr = inR ? 1.0f : anyr;
      const float pv = inR ? mv[k] : 0.0f;
      if (pv > bpV) { bpV = pv; bpI = jj; }
      const float nv = kd2 + (inR ? 1000.0f : 0.0f);
      if (nv < bnV) { bnV = nv; bnI = jj; }
      const float oq = inR ? 0.0f : ov[k];
      if (oq > boV) { boV = oq; boI = jj; }
    }
  }

  // combine the two lanes of each row (first-index tie-break = jnp semantics)
  {
    float v2; int i2; bool take;
    v2 = __shfl_xor(bpV, 1, 32); i2 = __shfl_xor(bpI, 1, 32);
    take = (v2 > bpV) || (v2 == bpV && i2 < bpI);
    bpV = take ? v2 : bpV; bpI = take ? i2 : bpI;
    v2 = __shfl_xor(bnV, 1, 32); i2 = __shfl_xor(bnI, 1, 32);
    take = (v2 < bnV) || (v2 == bnV && i2 < bnI);
    bnV = take ? v2 : bnV; bnI = take ? i2 : bnI;
    v2 = __shfl_xor(boV, 1, 32); i2 = __shfl_xor(boI, 1, 32);
    take = (v2 > boV) || (v2 == boV && i2 < boI);
    boV = take ? v2 : boV; boI = take ? i2 : boI;
    anyr = fmaxf(anyr, __shfl_xor(anyr, 1, 32));
  }
  if (h == 0) {
    s_idx[wave][r][0] = bpI;
    s_idx[wave][r][1] = bnI;
    s_idx[wave][r][2] = boI;
    s_any[wave][r] = anyr;
  }
  wave_lds_fence();   // wave-local: s_idx/s_any private to this wave

  // gathered descriptor distances: 3 columns x 64 channels per row,
  // 32 lanes x 2 channels each, full-wave butterfly reduce.
  const float* ad = adc + (size_t)b * CC * MM;
  const float* pd = pdc + (size_t)b * CC * MM;
  for (int rr = 0; rr < 16; ++rr) {
    const int i = i0 + rr;
    const float a0 = ad[(size_t)lane * MM + i];
    const float a1 = ad[(size_t)(lane + 32) * MM + i];
    const int jp = s_idx[wave][rr][0];
    const int jn = s_idx[wave][rr][1];
    const int jo = s_idx[wave][rr][2];
    const float dp0 = a0 - pd[(size_t)lane * MM + jp];
    const float dp1 = a1 - pd[(size_t)(lane + 32) * MM + jp];
    const float dn0 = a0 - pd[(size_t)lane * MM + jn];
    const float dn1 = a1 - pd[(size_t)(lane + 32) * MM + jn];
    const float do0 = a0 - pd[(size_t)lane * MM + jo];
    const float do1 = a1 - pd[(size_t)(lane + 32) * MM + jo];
    const float sp = wave_sum32(dp0 * dp0 + dp1 * dp1);
    const float sn = wave_sum32(dn0 * dn0 + dn1 * dn1);
    const float so = wave_sum32(do0 * do0 + do1 * do1);
    if (lane == 0) {
      const float dpd = sqrtf(sp);
      const float dfc = sqrtf(sn);
      const float dor = sqrtf(so);
      const float sel = (rs[(size_t)b * MM + i] < 0.5f) ? 1.0f : 0.0f;
      const float nd  = sel * dfc + (1.0f - sel) * dor;
      const float any = s_any[wave][rr];
      bcl [(size_t)b * MM + i] = (dpd - nd + 0.5f) * any;  // TRIPLE_LOSS_GAMMA
      anyv[(size_t)b * MM + i] = any;
    }
  }
}

// ---------------------------------------------------------------------------
// K2: deterministic per-batch reductions (pos_count, active count, sum w).
// ---------------------------------------------------------------------------
__global__ __launch_bounds__(256)
void k2_batchred(const float* __restrict__ bcl, const float* __restrict__ anyv,
                 const float* __restrict__ sig, float* __restrict__ acc)
{
  __shared__ float sp[256], sa[256], sw[256];
  const int b = blockIdx.x, t = threadIdx.x;
  float p = 0.f, aq = 0.f, w = 0.f;
  for (int i = t; i < MM; i += 256) {
    p  += anyv[(size_t)b * MM + i];
    aq += (bcl[(size_t)b * MM + i] > 1e-5f) ? 1.f : 0.f;
    w  += fmaxf(1.0f - sig[(size_t)b * MM + i], 0.0f);   // SIGMA_MAX = 1.0
  }
  sp[t] = p; sa[t] = aq; sw[t] = w;
  __syncthreads();
  for (int s = 128; s > 0; s >>= 1) {
    if (t < s) { sp[t] += sp[t + s]; sa[t] += sa[t + s]; sw[t] += sw[t + s]; }
    __syncthreads();
  }
  if (t == 0) { acc[b] = sp[0]; acc[BB + b] = sa[0]; acc[2 * BB + b] = sw[0]; }
}

// ---------------------------------------------------------------------------
// K3: final loss + active_percentage.
// ---------------------------------------------------------------------------
__global__ __launch_bounds__(256)
void k3_final(const float* __restrict__ bcl, const float* __restrict__ sig,
              const float* __restrict__ acc, float* __restrict__ out)
{
  const int idx = blockIdx.x * 256 + threadIdx.x;
  if (idx >= BB * MM) return;
  const int b = idx >> 10;                          // MM = 1024
  const float pc      = acc[b];
  const float scaling = (float)MM / (pc + 1.0f);
  const float meanw   = acc[2 * BB + b] * (1.0f / (float)MM);
  const float w       = fmaxf(1.0f - sig[idx], 0.0f) / meanw;
  out[idx] = w * fmaxf(bcl[idx], 0.0f) * scaling;
  if ((idx & (MM - 1)) == 0)
    out[BB * MM + b] = acc[BB + b] / (pc + 1.0f);   // active_percentage
}

extern "C" void kernel_launch(void* const* d_in, const int* /*in_sizes*/, int /*n_in*/,
                              void* d_out, int /*out_size*/, void* d_ws, size_t /*ws_size*/,
                              hipStream_t stream) {
  const float* akp = (const float*)d_in[0];  // B,3,M
  const float* adc = (const float*)d_in[1];  // B,C,M
  const float* pkp = (const float*)d_in[2];  // B,3,M
  const float* pdc = (const float*)d_in[3];  // B,C,M
  const float* sig = (const float*)d_in[4];  // B,M
  const float* rm  = (const float*)d_in[5];  // B,M,M
  const float* ro  = (const float*)d_in[6];  // B,M,M
  const float* rs  = (const float*)d_in[7];  // B,M
  float* out = (float*)d_out;                // loss (B*M) then active_percentage (B)

  float* ws   = (float*)d_ws;
  float* bcl  = ws;                 // B*M
  float* anyv = ws + BB * MM;       // B*M
  float* acc  = ws + 2 * BB * MM;   // 3*B

  k1_select<<<dim3(BB * MM / 64), dim3(128), 0, stream>>>(akp, adc, pkp, pdc,
                                                          rs, rm, ro, bcl, anyv);
  k2_batchred<<<dim3(BB), dim3(256), 0, stream>>>(bcl, anyv, sig, acc);
  k3_final<<<dim3((BB * MM + 255) / 256), dim3(256), 0, stream>>>(bcl, sig, acc, out);
}